// CANMultiHeadMessagePassing_66760971649670
// MI455X (gfx1250) — compile-verified
//
#include <hip/hip_runtime.h>
#include <hip/hip_bf16.h>
#include <stdint.h>

#define N_NODES 50000
#define N_EDGES 1600000
#define IN_CH   128
#define OUT_CH  64
#define N_HEADS 4
#define XCH     (N_HEADS * OUT_CH)   // 256
#define M_TILES (N_NODES / 16)       // 3125 (exact)

typedef __attribute__((ext_vector_type(2))) float v2f;
typedef __attribute__((ext_vector_type(8))) float v8f;

// ---------------------------------------------------------------------------
// Order-preserving float <-> uint mapping for atomicMax-based segment_max.
// ---------------------------------------------------------------------------
__device__ __forceinline__ unsigned ordmap(float f) {
    unsigned u = __float_as_uint(f);
    return (u & 0x80000000u) ? ~u : (u | 0x80000000u);
}
__device__ __forceinline__ float ordunmap(unsigned u) {
    unsigned b = (u & 0x80000000u) ? (u ^ 0x80000000u) : ~u;
    return __uint_as_float(b);
}

// ---------------------------------------------------------------------------
// Kernel 1: x_msg[n, h*64+o] = sum_i X[n,i] * W[h,i,o] via V_WMMA_F32_16X16X4_F32.
// One wave per (16-row m-tile, head); wave computes a 16x64 strip as 4 16x16
// subtiles, reusing the A fragment across subtiles. fp32 in/out (exact).
// A 16x4 frag:  lanes 0-15: M=lane, v0=K0,v1=K1 ; lanes 16-31: M=lane-16, v0=K2,v1=K3
// B 4x16 frag:  lanes 0-15: N=lane, v0=K0,v1=K1 ; lanes 16-31: N=lane-16, v0=K2,v1=K3
// C/D 16x16:    VGPR r: lanes 0-15 -> M=r, lanes 16-31 -> M=r+8, N=lane%16
// ---------------------------------------------------------------------------
__global__ __launch_bounds__(256) void gemm_xmsg_wmma(
    const float* __restrict__ X,   // [N_NODES, 128]
    const float* __restrict__ W,   // [4, 128, 64]
    float* __restrict__ xmsg)      // [N_NODES, 256]
{
    const int wave = threadIdx.x >> 5;
    const int lane = threadIdx.x & 31;
    const int gw   = blockIdx.x * 8 + wave;          // global wave id
    if (gw >= M_TILES * N_HEADS) return;             // wave-uniform: EXEC stays full
    const int mt = gw >> 2;                          // m-tile
    const int h  = gw & 3;                           // head
    const int m0 = mt * 16;
    const int nl = lane & 15;
    const int hi = (lane >> 4) & 1;                  // lane half selects K pair

    v8f acc[4];
#pragma unroll
    for (int nt = 0; nt < 4; ++nt) acc[nt] = (v8f){0.f,0.f,0.f,0.f,0.f,0.f,0.f,0.f};

    const float* wh = W + (size_t)h * IN_CH * OUT_CH;

    for (int k0 = 0; k0 < IN_CH; k0 += 4) {
        const int ka = k0 + 2 * hi;
        // A fragment (shared by all 4 N-subtiles)
        const float* ap = X + (size_t)(m0 + nl) * IN_CH + ka;
        v2f a; a.x = ap[0]; a.y = ap[1];
#pragma unroll
        for (int nt = 0; nt < 4; ++nt) {
            const float* bp = wh + (size_t)ka * OUT_CH + (nt * 16 + nl);
            v2f b; b.x = bp[0]; b.y = bp[OUT_CH];   // rows ka, ka+1
            acc[nt] = __builtin_amdgcn_wmma_f32_16x16x4_f32(
                false, a, false, b, (short)0, acc[nt], false, false);
        }
    }

#pragma unroll
    for (int nt = 0; nt < 4; ++nt) {
#pragma unroll
        for (int r = 0; r < 8; ++r) {
            const int m = m0 + r + 8 * hi;
            const int c = h * OUT_CH + nt * 16 + nl;
            xmsg[(size_t)m * XCH + c] = acc[nt][r];
        }
    }
}

// ---------------------------------------------------------------------------
// Kernel 2: per-(node, head) attention scalars s, t (64-wide dots).
// ---------------------------------------------------------------------------
__global__ __launch_bounds__(256) void node_logits(
    const float* __restrict__ xmsg, const float* __restrict__ att, // att: [4, 128]
    float* __restrict__ s, float* __restrict__ t)
{
    const int idx = blockIdx.x * blockDim.x + threadIdx.x;   // node*4 + h
    if (idx >= N_NODES * N_HEADS) return;
    const int n = idx >> 2, h = idx & 3;
    const float4* v4  = (const float4*)(xmsg + (size_t)n * XCH + h * OUT_CH);
    const float4* as4 = (const float4*)(att + h * 2 * OUT_CH);
    const float4* at4 = (const float4*)(att + h * 2 * OUT_CH + OUT_CH);
    float ss = 0.f, tt = 0.f;
#pragma unroll
    for (int i = 0; i < 16; ++i) {
        float4 xv = v4[i], a = as4[i], b = at4[i];
        ss += xv.x*a.x + xv.y*a.y + xv.z*a.z + xv.w*a.w;
        tt += xv.x*b.x + xv.y*b.y + xv.z*b.z + xv.w*b.w;
    }
    s[idx] = ss; t[idx] = tt;
}

// ---------------------------------------------------------------------------
// Kernel 3: init row_max to mapped(-inf).
// ---------------------------------------------------------------------------
__global__ void init_rmax(unsigned* __restrict__ rmax)
{
    const int idx = blockIdx.x * blockDim.x + threadIdx.x;
    if (idx < N_NODES * N_HEADS) rmax[idx] = 0x007FFFFFu;   // ordmap(-inf)
}

// ---------------------------------------------------------------------------
// Kernel 4: edge logits + leaky_relu + edge_vals + segment_max (atomicMax u32).
// ---------------------------------------------------------------------------
__global__ __launch_bounds__(256) void edge_logits_max(
    const float* __restrict__ s, const float* __restrict__ t,
    const int* __restrict__ esrc, const int* __restrict__ etgt,
    const float* __restrict__ evals,
    float* __restrict__ ebuf, unsigned* __restrict__ rmax)
{
    const int idx = blockIdx.x * blockDim.x + threadIdx.x;   // edge*4 + h
    if (idx >= N_EDGES * N_HEADS) return;
    const int e = idx >> 2, h = idx & 3;
    const int src = esrc[e], tgt = etgt[e];
    float v = s[src * 4 + h] + t[tgt * 4 + h];
    v = (v > 0.f) ? v : 0.01f * v;          // leaky_relu(0.01)
    v *= evals[e];
    ebuf[idx] = v;
    atomicMax(rmax + tgt * 4 + h, ordmap(v));
}

// ---------------------------------------------------------------------------
// Kernel 5: e_exp = exp(e - row_max[tgt]); segment_sum via atomicAdd f32.
// ---------------------------------------------------------------------------
__global__ __launch_bounds__(256) void edge_exp_sum(
    const int* __restrict__ etgt, const unsigned* __restrict__ rmax,
    float* __restrict__ ebuf, float* __restrict__ rsum)
{
    const int idx = blockIdx.x * blockDim.x + threadIdx.x;
    if (idx >= N_EDGES * N_HEADS) return;
    const int e = idx >> 2, h = idx & 3;
    const int tgt = etgt[e];
    const float m  = ordunmap(rmax[tgt * 4 + h]);
    const float ee = expf(ebuf[idx] - m);
    ebuf[idx] = ee;
    atomicAdd(rsum + tgt * 4 + h, ee);
}

// ---------------------------------------------------------------------------
// Kernel 6: reciprocal of row sums (0 for isolated nodes).
// ---------------------------------------------------------------------------
__global__ void row_inv(const float* __restrict__ rsum, float* __restrict__ inv)
{
    const int idx = blockIdx.x * blockDim.x + threadIdx.x;
    if (idx >= N_NODES * N_HEADS) return;
    const float sv = rsum[idx];
    inv[idx] = (sv > 0.f) ? (1.f / sv) : 0.f;
}

// ---------------------------------------------------------------------------
// Kernel 7: scatter out[tgt] += alpha * x_msg[src]. One wave32 per edge;
// lane owns 8 contiguous channels (head = lane/8). x_msg (51 MB) and out
// (51 MB) both live in the 192 MB L2, so gathers + atomics resolve at L2.
// ---------------------------------------------------------------------------
__global__ __launch_bounds__(256) void scatter_out(
    const float* __restrict__ xmsg,
    const int* __restrict__ esrc, const int* __restrict__ etgt,
    const float* __restrict__ ebuf, const float* __restrict__ inv,
    float* __restrict__ out)
{
    const int e = blockIdx.x * 8 + (threadIdx.x >> 5);
    if (e >= N_EDGES) return;
    const int lane = threadIdx.x & 31;
    const int src = esrc[e], tgt = etgt[e];
    const int h = lane >> 3;                          // 8 lanes per head (64 ch)
    const float alpha = ebuf[e * 4 + h] * inv[tgt * 4 + h];

    const float4* p = (const float4*)(xmsg + (size_t)src * XCH) + lane * 2;
    const float4 a = p[0], b = p[1];
    float* q = out + (size_t)tgt * XCH + lane * 8;
    atomicAdd(q + 0, alpha * a.x);
    atomicAdd(q + 1, alpha * a.y);
    atomicAdd(q + 2, alpha * a.z);
    atomicAdd(q + 3, alpha * a.w);
    atomicAdd(q + 4, alpha * b.x);
    atomicAdd(q + 5, alpha * b.y);
    atomicAdd(q + 6, alpha * b.z);
    atomicAdd(q + 7, alpha * b.w);
}

// ---------------------------------------------------------------------------
extern "C" void kernel_launch(void* const* d_in, const int* in_sizes, int n_in,
                              void* d_out, int out_size, void* d_ws, size_t ws_size,
                              hipStream_t stream)
{
    const float* x     = (const float*)d_in[0];   // [50000,128]
    const int*   etgt  = (const int*)  d_in[1];   // [1.6M]
    const int*   esrc  = (const int*)  d_in[2];   // [1.6M]
    const float* evals = (const float*)d_in[3];   // [1.6M]
    const float* w     = (const float*)d_in[4];   // [4,128,64]
    const float* att   = (const float*)d_in[5];   // [4,128]
    float*       out   = (float*)d_out;           // [50000,256]

    // Workspace carving (256B aligned slabs), total ~81 MB.
    char* ws = (char*)d_ws;
    auto take = [&](size_t bytes) { char* p = ws; ws += (bytes + 255) & ~(size_t)255; return p; };
    float*    xmsg = (float*)   take((size_t)N_NODES * XCH * sizeof(float));     // 51.2 MB
    float*    sarr = (float*)   take((size_t)N_NODES * N_HEADS * sizeof(float));
    float*    tarr = (float*)   take((size_t)N_NODES * N_HEADS * sizeof(float));
    float*    ebuf = (float*)   take((size_t)N_EDGES * N_HEADS * sizeof(float)); // 25.6 MB
    unsigned* rmax = (unsigned*)take((size_t)N_NODES * N_HEADS * sizeof(unsigned));
    float*    rsum = (float*)   take((size_t)N_NODES * N_HEADS * sizeof(float));
    float*    inv  = (float*)   take((size_t)N_NODES * N_HEADS * sizeof(float));

    // Zero / -inf initialization (deterministic each call; capture-safe).
    hipMemsetAsync(out, 0, (size_t)N_NODES * XCH * sizeof(float), stream);
    hipMemsetAsync(rsum, 0, (size_t)N_NODES * N_HEADS * sizeof(float), stream);
    init_rmax<<<(N_NODES * N_HEADS + 255) / 256, 256, 0, stream>>>(rmax);

    // 1) GEMM via f32 WMMA: 3125 m-tiles x 4 heads = 12500 waves, 8 waves/block.
    gemm_xmsg_wmma<<<(M_TILES * N_HEADS + 7) / 8, 256, 0, stream>>>(x, w, xmsg);

    // 2) Per-node attention scalars.
    node_logits<<<(N_NODES * N_HEADS + 255) / 256, 256, 0, stream>>>(xmsg, att, sarr, tarr);

    // 3) Edge logits + segment max.
    edge_logits_max<<<(N_EDGES * N_HEADS + 255) / 256, 256, 0, stream>>>(
        sarr, tarr, esrc, etgt, evals, ebuf, rmax);

    // 4) exp + segment sum.
    edge_exp_sum<<<(N_EDGES * N_HEADS + 255) / 256, 256, 0, stream>>>(etgt, rmax, ebuf, rsum);

    // 5) 1/rowsum.
    row_inv<<<(N_NODES * N_HEADS + 255) / 256, 256, 0, stream>>>(rsum, inv);

    // 6) Weighted scatter-add: one wave per edge, 8 waves/block.
    scatter_out<<<(N_EDGES + 7) / 8, 256, 0, stream>>>(xmsg, esrc, etgt, ebuf, inv, out);
}